// GAT_64785286692910
// MI455X (gfx1250) — compile-verified
//
#include <hip/hip_runtime.h>
#include <math.h>

typedef float v2f __attribute__((ext_vector_type(2)));
typedef float v8f __attribute__((ext_vector_type(8)));

__device__ __forceinline__ unsigned fenc(float f) {
  unsigned b = __float_as_uint(f);
  return (b & 0x80000000u) ? ~b : (b | 0x80000000u);
}
__device__ __forceinline__ float fdec(unsigned u) {
  return __uint_as_float((u & 0x80000000u) ? (u & 0x7FFFFFFFu) : ~u);
}
__device__ __forceinline__ float lrelu(float v) { return v > 0.f ? v : 0.2f * v; }
__device__ __forceinline__ float eluf(float v)  { return v > 0.f ? v : expm1f(v); }

// ---------------------------------------------------------------- utilities
__global__ void k_zero(float* __restrict__ p, long n) {
  long i = (long)blockIdx.x * blockDim.x + threadIdx.x;
  if (i < n) p[i] = 0.f;
}

// K=2 input projections: xl1, xr1, and H init = bias1 + skip
__global__ void k_node_lin1(const float* __restrict__ x,
                            const float* __restrict__ Wl, const float* __restrict__ bl,
                            const float* __restrict__ Wr, const float* __restrict__ br,
                            const float* __restrict__ Ws, const float* __restrict__ bs,
                            const float* __restrict__ bias1,
                            float* __restrict__ xl, float* __restrict__ xr,
                            float* __restrict__ H, int N) {
  long i = (long)blockIdx.x * blockDim.x + threadIdx.x;
  if (i >= (long)N * 64) return;
  int n = (int)(i >> 6), c = (int)(i & 63);
  float x0 = x[2 * n], x1 = x[2 * n + 1];
  xl[i] = x0 * Wl[c] + x1 * Wl[64 + c] + bl[c];
  xr[i] = x0 * Wr[c] + x1 * Wr[64 + c] + br[c];
  H[i]  = x0 * Ws[c] + x1 * Ws[64 + c] + bs[c] + bias1[c];
}

// degree + incoming edge-attr sum (for self-loop fill_value='mean')
__global__ void k_deg(const int* __restrict__ dst0, const float* __restrict__ ea,
                      float* __restrict__ deg, float* __restrict__ loopsum, int E) {
  int e = blockIdx.x * blockDim.x + threadIdx.x;
  if (e >= E) return;
  int d = dst0[e];
  atomicAdd(&deg[d], 1.f);
  atomicAdd(&loopsum[d], ea[e]);
}
__global__ void k_loopdiv(float* __restrict__ loopa, const float* __restrict__ deg, int N) {
  int n = blockIdx.x * blockDim.x + threadIdx.x;
  if (n >= N) return;
  loopa[n] = loopa[n] / fmaxf(deg[n], 1.f);
}

// ------------------------------------------------- layer 1 (H=2, C=32) edges
__global__ void k_edge_logit1(const int* __restrict__ src0, const int* __restrict__ dst0,
                              const float* __restrict__ ea, const float* __restrict__ loopa,
                              const float* __restrict__ xl, const float* __restrict__ xr,
                              const float* __restrict__ We, const float* __restrict__ att,
                              float* __restrict__ logit, unsigned* __restrict__ amax,
                              int E, int Et) {
  int w = blockIdx.x * (blockDim.x >> 5) + (threadIdx.x >> 5);
  int lane = threadIdx.x & 31;
  if (w >= Et) return;
  int s, d; float a;
  if (w < E) { s = src0[w]; d = dst0[w]; a = ea[w]; }
  else       { s = d = w - E; a = loopa[w - E]; }
  long so = (long)s * 64, dofs = (long)d * 64;
  float m0 = lrelu(xl[so + lane]      + xr[dofs + lane]      + a * We[lane]);
  float m1 = lrelu(xl[so + lane + 32] + xr[dofs + lane + 32] + a * We[lane + 32]);
  float p0 = m0 * att[lane];
  float p1 = m1 * att[lane + 32];
  for (int off = 16; off > 0; off >>= 1) {
    p0 += __shfl_down(p0, off);
    p1 += __shfl_down(p1, off);
  }
  if (lane == 0) {
    logit[2 * w]     = p0;
    logit[2 * w + 1] = p1;
    atomicMax(&amax[2 * d],     fenc(p0));
    atomicMax(&amax[2 * d + 1], fenc(p1));
  }
}

__global__ void k_edge_exp1(const int* __restrict__ dst0, float* __restrict__ logit,
                            const unsigned* __restrict__ amax, float* __restrict__ denom,
                            int E, int Et) {
  int i = blockIdx.x * blockDim.x + threadIdx.x;
  if (i >= 2 * Et) return;
  int e = i >> 1, h = i & 1;
  int d = (e < E) ? dst0[e] : (e - E);
  float v = __expf(logit[i] - fdec(amax[2 * d + h]));
  logit[i] = v;
  atomicAdd(&denom[2 * d + h], v);
}

__global__ void k_edge_aggr1(const int* __restrict__ src0, const int* __restrict__ dst0,
                             const float* __restrict__ xl, const float* __restrict__ expv,
                             const float* __restrict__ denom, float* __restrict__ H,
                             int E, int Et) {
  int w = blockIdx.x * (blockDim.x >> 5) + (threadIdx.x >> 5);
  int lane = threadIdx.x & 31;
  if (w >= Et) return;
  int s, d;
  if (w < E) { s = src0[w]; d = dst0[w]; } else { s = d = w - E; }
  float a0 = expv[2 * w]     / (denom[2 * d]     + 1e-16f);
  float a1 = expv[2 * w + 1] / (denom[2 * d + 1] + 1e-16f);
  long so = (long)s * 64, dofs = (long)d * 64;
  atomicAdd(&H[dofs + lane],      xl[so + lane]      * a0);
  atomicAdd(&H[dofs + lane + 32], xl[so + lane + 32] * a1);
}

// ------------------------------------------------------------- batch norm
__global__ void k_bn_stats(const float* __restrict__ H, float* __restrict__ stats, int N) {
  __shared__ float ss[64], sq[64];
  int tid = threadIdx.x;
  int c = tid & 63;
  if (tid < 64) { ss[tid] = 0.f; sq[tid] = 0.f; }
  __syncthreads();
  float s = 0.f, q = 0.f;
  for (int r = blockIdx.x * 4 + (tid >> 6); r < N; r += gridDim.x * 4) {
    float v = H[(long)r * 64 + c];
    s += v; q += v * v;
  }
  atomicAdd(&ss[c], s);
  atomicAdd(&sq[c], q);
  __syncthreads();
  if (tid < 64) {
    atomicAdd(&stats[tid], ss[tid]);
    atomicAdd(&stats[64 + tid], sq[tid]);
  }
}

__global__ void k_bn_apply(float* __restrict__ H, const float* __restrict__ stats,
                           const float* __restrict__ g, const float* __restrict__ b, int N) {
  long i = (long)blockIdx.x * blockDim.x + threadIdx.x;
  if (i >= (long)N * 64) return;
  int c = (int)(i & 63);
  float invN = 1.f / (float)N;
  float mu = stats[c] * invN;
  float var = stats[64 + c] * invN - mu * mu;
  float v = (H[i] - mu) * rsqrtf(var + 1e-5f) * g[c] + b[c];
  H[i] = eluf(v);
}

// ----------------------------------- WMMA dual-output GEMM: [N,K] @ [K,32] x2
// one wave handles 16 rows; 4 f32 16x16x4 WMMA accumulators (2 weights x 2 col tiles)
__global__ void k_gemm16_dual(const float* __restrict__ A, int lda, int K,
                              const float* __restrict__ W1, const float* __restrict__ b1,
                              float* __restrict__ O1,
                              const float* __restrict__ W2, const float* __restrict__ b2,
                              float* __restrict__ O2,
                              int N, int relu) {
  int wave = blockIdx.x * (blockDim.x >> 5) + (threadIdx.x >> 5);
  int lane = threadIdx.x & 31;
  int ntiles = (N + 15) >> 4;
  if (wave >= ntiles) return;
  int row0 = wave * 16;
  int low = lane & 15, hi = lane >> 4;
  int arow = row0 + low;
  if (arow > N - 1) arow = N - 1;   // clamp (discarded by store guard)
  const float* Ar = A + (long)arow * lda;
  v8f acc0 = {}, acc1 = {}, acc2 = {}, acc3 = {};
  for (int k0 = 0; k0 < K; k0 += 4) {
    int ka = k0 + 2 * hi;
    v2f a; a.x = Ar[ka]; a.y = Ar[ka + 1];
    v2f bA, bB, bC, bD;
    bA.x = W1[ka * 32 + low];            bA.y = W1[(ka + 1) * 32 + low];
    bB.x = W1[ka * 32 + 16 + low];       bB.y = W1[(ka + 1) * 32 + 16 + low];
    bC.x = W2[ka * 32 + low];            bC.y = W2[(ka + 1) * 32 + low];
    bD.x = W2[ka * 32 + 16 + low];       bD.y = W2[(ka + 1) * 32 + 16 + low];
    acc0 = __builtin_amdgcn_wmma_f32_16x16x4_f32(false, a, false, bA, (short)0, acc0, false, false);
    acc1 = __builtin_amdgcn_wmma_f32_16x16x4_f32(false, a, false, bB, (short)0, acc1, false, false);
    acc2 = __builtin_amdgcn_wmma_f32_16x16x4_f32(false, a, false, bC, (short)0, acc2, false, false);
    acc3 = __builtin_amdgcn_wmma_f32_16x16x4_f32(false, a, false, bD, (short)0, acc3, false, false);
  }
#pragma unroll
  for (int v = 0; v < 8; ++v) {
    int r = row0 + v + 8 * hi;
    if (r < N) {
      float o0 = acc0[v] + b1[low];
      float o1 = acc1[v] + b1[16 + low];
      float o2 = acc2[v] + b2[low];
      float o3 = acc3[v] + b2[16 + low];
      if (relu) { o0 = fmaxf(o0, 0.f); o1 = fmaxf(o1, 0.f); o2 = fmaxf(o2, 0.f); o3 = fmaxf(o3, 0.f); }
      O1[(long)r * 32 + low]      = o0;
      O1[(long)r * 32 + 16 + low] = o1;
      O2[(long)r * 32 + low]      = o2;
      O2[(long)r * 32 + 16 + low] = o3;
    }
  }
}

// ------------------------------------------------- layer 2 (H=1, C=32) edges
__global__ void k_edge_logit2(const int* __restrict__ src0, const int* __restrict__ dst0,
                              const float* __restrict__ ea, const float* __restrict__ loopa,
                              const float* __restrict__ xl, const float* __restrict__ xr,
                              const float* __restrict__ We, const float* __restrict__ att,
                              float* __restrict__ logit, unsigned* __restrict__ amax,
                              int E, int Et) {
  int w = blockIdx.x * (blockDim.x >> 5) + (threadIdx.x >> 5);
  int lane = threadIdx.x & 31;
  if (w >= Et) return;
  int s, d; float a;
  if (w < E) { s = src0[w]; d = dst0[w]; a = ea[w]; }
  else       { s = d = w - E; a = loopa[w - E]; }
  float m = lrelu(xl[(long)s * 32 + lane] + xr[(long)d * 32 + lane] + a * We[lane]);
  float p = m * att[lane];
  for (int off = 16; off > 0; off >>= 1) p += __shfl_down(p, off);
  if (lane == 0) {
    logit[w] = p;
    atomicMax(&amax[d], fenc(p));
  }
}

__global__ void k_edge_exp2(const int* __restrict__ dst0, float* __restrict__ logit,
                            const unsigned* __restrict__ amax, float* __restrict__ denom,
                            int E, int Et) {
  int e = blockIdx.x * blockDim.x + threadIdx.x;
  if (e >= Et) return;
  int d = (e < E) ? dst0[e] : (e - E);
  float v = __expf(logit[e] - fdec(amax[d]));
  logit[e] = v;
  atomicAdd(&denom[d], v);
}

__global__ void k_edge_aggr2(const int* __restrict__ src0, const int* __restrict__ dst0,
                             const float* __restrict__ xl, const float* __restrict__ expv,
                             const float* __restrict__ denom, float* __restrict__ H2,
                             float* __restrict__ alpha_out, int E, int Et) {
  int w = blockIdx.x * (blockDim.x >> 5) + (threadIdx.x >> 5);
  int lane = threadIdx.x & 31;
  if (w >= Et) return;
  int s, d;
  if (w < E) { s = src0[w]; d = dst0[w]; } else { s = d = w - E; }
  float a = expv[w] / (denom[d] + 1e-16f);
  if (lane == 0) alpha_out[w] = a;
  atomicAdd(&H2[(long)d * 32 + lane], xl[(long)s * 32 + lane] * a);
}

__global__ void k_final2(float* __restrict__ H2, const float* __restrict__ bias2, int N) {
  long i = (long)blockIdx.x * blockDim.x + threadIdx.x;
  if (i >= (long)N * 32) return;
  H2[i] = eluf(H2[i] + bias2[i & 31]);
}

// -------------------------------------------- classifier heads + log_softmax
__global__ void k_classify(const float* __restrict__ Tt, const float* __restrict__ Tc,
                           const float* __restrict__ Wt2, const float* __restrict__ bt2,
                           const float* __restrict__ Wc2, const float* __restrict__ bc2,
                           float* __restrict__ out, int N) {
  int n = blockIdx.x * blockDim.x + threadIdx.x;
  if (n >= N) return;
  const float* rt = Tt + (long)n * 32;
  const float* rc = Tc + (long)n * 32;
  float zt[20], zc[10];
#pragma unroll
  for (int k = 0; k < 20; ++k) {
    float s = bt2[k];
#pragma unroll
    for (int j = 0; j < 32; ++j) s += rt[j] * Wt2[j * 20 + k];
    zt[k] = s;
  }
#pragma unroll
  for (int k = 0; k < 10; ++k) {
    float s = bc2[k];
#pragma unroll
    for (int j = 0; j < 32; ++j) s += rc[j] * Wc2[j * 10 + k];
    zc[k] = s;
  }
  float mt = zt[0], mc = zc[0];
#pragma unroll
  for (int k = 1; k < 20; ++k) mt = fmaxf(mt, zt[k]);
#pragma unroll
  for (int k = 1; k < 10; ++k) mc = fmaxf(mc, zc[k]);
  float st = 0.f, sc = 0.f;
#pragma unroll
  for (int k = 0; k < 20; ++k) st += expf(zt[k] - mt);
#pragma unroll
  for (int k = 0; k < 10; ++k) sc += expf(zc[k] - mc);
  float lt = logf(st), lc = logf(sc);
  float* o = out + (long)n * 30;
#pragma unroll
  for (int k = 0; k < 10; ++k) o[k] = zc[k] - mc - lc;
#pragma unroll
  for (int k = 0; k < 20; ++k) o[10 + k] = zt[k] - mt - lt;
}

// ---------------------------------------------------------------------------
extern "C" void kernel_launch(void* const* d_in, const int* in_sizes, int n_in,
                              void* d_out, int out_size, void* d_ws, size_t ws_size,
                              hipStream_t stream) {
  const float* x    = (const float*)d_in[0];
  const int*   ei   = (const int*)  d_in[1];
  const float* ea   = (const float*)d_in[2];
  const float* Wl1  = (const float*)d_in[3];
  const float* bl1  = (const float*)d_in[4];
  const float* Wr1  = (const float*)d_in[5];
  const float* br1  = (const float*)d_in[6];
  const float* We1  = (const float*)d_in[7];
  const float* att1 = (const float*)d_in[8];
  const float* bias1= (const float*)d_in[9];
  const float* skW  = (const float*)d_in[10];
  const float* skb  = (const float*)d_in[11];
  const float* bng  = (const float*)d_in[12];
  const float* bnb  = (const float*)d_in[13];
  const float* Wl2  = (const float*)d_in[14];
  const float* bl2  = (const float*)d_in[15];
  const float* Wr2  = (const float*)d_in[16];
  const float* br2  = (const float*)d_in[17];
  const float* We2  = (const float*)d_in[18];
  const float* att2 = (const float*)d_in[19];
  const float* bias2= (const float*)d_in[20];
  const float* Wc1  = (const float*)d_in[21];
  const float* bc1  = (const float*)d_in[22];
  const float* Wc2  = (const float*)d_in[23];
  const float* bc2  = (const float*)d_in[24];
  const float* Wt1  = (const float*)d_in[25];
  const float* bt1  = (const float*)d_in[26];
  const float* Wt2  = (const float*)d_in[27];
  const float* bt2  = (const float*)d_in[28];

  const int N  = in_sizes[0] / 2;
  const int E  = in_sizes[2];
  const int Et = E + N;
  const int* src0 = ei;
  const int* dst0 = ei + E;

  // workspace layout (floats)
  float* base   = (float*)d_ws;
  float* xl1    = base;                   // 64N
  float* xr1    = base + 64L * N;         // 64N (later: Tt / Tc)
  float* H      = base + 128L * N;        // 64N (later lower half: H2)
  float* logit  = base + 192L * N;        // 2*Et (later: logit2)
  float* amax_f = logit + 2L * Et;        // 2N  (later: amax2)
  float* denom  = amax_f + 2L * N;        // 2N  (later: denom2)
  float* deg    = denom + 2L * N;         // N
  float* loopa  = deg + N;                // N
  float* stats  = loopa + N;              // 128
  unsigned* amax = (unsigned*)amax_f;

  float* xl2 = xl1;                       // reuse: 32N
  float* xr2 = xl1 + 32L * N;             // reuse: 32N
  float* Tt  = xr1;                       // reuse: 32N
  float* Tc  = xr1 + 32L * N;             // reuse: 32N
  float* H2  = H;                         // reuse: 32N
  float* logit2 = logit;
  unsigned* amax2 = amax;
  float* denom2 = denom;

  float* out = (float*)d_out;
  float* alpha_out = out + 30L * N;

  const int B = 256;
  const int WPB = B / 32;  // waves per block
  auto gr = [](long n, int b) { return (int)((n + b - 1) / b); };

  // zero: amax, denom, deg, loopa, stats  (contiguous: 6N + 128 floats)
  long z1 = 6L * N + 128;
  k_zero<<<gr(z1, B), B, 0, stream>>>(amax_f, z1);

  // layer-1 node projections + H init (bias1 + skip)
  k_node_lin1<<<gr(64L * N, B), B, 0, stream>>>(x, Wl1, bl1, Wr1, br1, skW, skb, bias1,
                                                xl1, xr1, H, N);
  // self-loop attr = mean of incoming edge attrs
  k_deg<<<gr(E, B), B, 0, stream>>>(dst0, ea, deg, loopa, E);
  k_loopdiv<<<gr(N, B), B, 0, stream>>>(loopa, deg, N);

  // layer 1 segment softmax + aggregation
  int eb = gr(Et, WPB);
  k_edge_logit1<<<eb, B, 0, stream>>>(src0, dst0, ea, loopa, xl1, xr1, We1, att1,
                                      logit, amax, E, Et);
  k_edge_exp1<<<gr(2L * Et, B), B, 0, stream>>>(dst0, logit, amax, denom, E, Et);
  k_edge_aggr1<<<eb, B, 0, stream>>>(src0, dst0, xl1, logit, denom, H, E, Et);

  // batch norm (training-style batch stats) + ELU
  k_bn_stats<<<512, B, 0, stream>>>(H, stats, N);
  k_bn_apply<<<gr(64L * N, B), B, 0, stream>>>(H, stats, bng, bnb, N);

  // layer-2 node projections via WMMA: [N,64] @ [64,32] (lin_l & lin_r fused)
  int gtiles = gr((N + 15) / 16, WPB);
  k_gemm16_dual<<<gtiles, B, 0, stream>>>(H, 64, 64, Wl2, bl2, xl2, Wr2, br2, xr2, N, 0);

  // re-zero layer-2 accumulators (after GEMM consumed H)
  k_zero<<<gr(32L * N, B), B, 0, stream>>>(H2, 32L * N);
  k_zero<<<gr(4L * N, B), B, 0, stream>>>(amax_f, 4L * N);

  // layer 2 segment softmax + aggregation (+ alpha output)
  k_edge_logit2<<<eb, B, 0, stream>>>(src0, dst0, ea, loopa, xl2, xr2, We2, att2,
                                      logit2, amax2, E, Et);
  k_edge_exp2<<<gr(Et, B), B, 0, stream>>>(dst0, logit2, amax2, denom2, E, Et);
  k_edge_aggr2<<<eb, B, 0, stream>>>(src0, dst0, xl2, logit2, denom2, H2, alpha_out, E, Et);
  k_final2<<<gr(32L * N, B), B, 0, stream>>>(H2, bias2, N);

  // classifier head GEMMs via WMMA: [N,32] @ [32,32] (type & clone fused, ReLU)
  k_gemm16_dual<<<gtiles, B, 0, stream>>>(H2, 32, 32, Wt1, bt1, Tt, Wc1, bc1, Tc, N, 1);

  // final matvecs + log_softmax -> d_out[N,30]
  k_classify<<<gr(N, B), B, 0, stream>>>(Tt, Tc, Wt2, bt2, Wc2, bc2, out, N);
}